// GAT_6949257085587
// MI455X (gfx1250) — compile-verified
//
#include <hip/hip_runtime.h>

// ---------------------------------------------------------------------------
// GAT layer for MI455X (gfx1250, wave32).
//
// Roofline: dominant traffic is entity_embs = 20000*64*64*4 B = 327.7 MB,
// streamed from HBM exactly once (15 us floor at 23.3 TB/s). Useful FLOPs
// ~0.66 GFLOP -> purely bandwidth bound, so precision stays f32 and each
// 16 KB node tile is staged once in LDS and reused for logits + weighted sum.
//
// Algebra: (X@W)@a == X@(W@a); a tiny pre-kernel computes wv1 = W@a1,
// wv2 = W@a2 (64 floats each, in d_ws) so Wh/We are never materialized.
//
// Matrix path: per-node logits e = tile(64x64) @ wv2 via
// V_WMMA_F32_16X16X4_F32, 4 M-tiles x 16 chained K-tiles, full f32 accum.
// B operand = wv2 replicated across all 16 N columns -> every lane holds the
// result; lanes 0/16 write it out per the documented 16x16 f32 C/D layout.
// Data movement uses CDNA5 async global->LDS (ASYNCcnt) when available.
// LDS tile uses a 68-dword row stride: bank = (4*row + k) % 64 -> conflict-
// free b64 A-fragment reads and conflict-free column reads in the epilogue.
// ---------------------------------------------------------------------------

#define NTOT    20000
#define MM      64
#define FF      64
#define STRIDE  68                      // padded LDS row stride in dwords
#define LALPHA  0.2f
#define NEG_BIG (-9000000000000000.0f)
#define WAVES   2                       // nodes (waves) per workgroup

typedef __attribute__((ext_vector_type(2))) float v2f;
typedef __attribute__((ext_vector_type(4))) float v4f;
typedef __attribute__((ext_vector_type(8))) float v8f;
typedef __attribute__((ext_vector_type(4))) int   v4i;

typedef __attribute__((address_space(1))) v4i* gv4i_p;  // global (AS1) int4*
typedef __attribute__((address_space(3))) v4i* lv4i_p;  // LDS (AS3) int4*

#if defined(__has_builtin)
#if __has_builtin(__builtin_amdgcn_global_load_async_to_lds_b128) && \
    __has_builtin(__builtin_amdgcn_s_wait_asynccnt)
#define USE_ASYNC 1
#endif
#endif
#ifndef USE_ASYNC
#define USE_ASYNC 0
#endif

// wv[0:64] = W @ a[0:64], wv[64:128] = W @ a[64:128]
__global__ void wv_precompute(const float* __restrict__ W,
                              const float* __restrict__ a,
                              float* __restrict__ wv) {
  const int f = threadIdx.x;            // 0..63
  float s1 = 0.f, s2 = 0.f;
#pragma unroll 8
  for (int h = 0; h < FF; ++h) {
    const float w = W[f * FF + h];
    s1 += w * a[h];
    s2 += w * a[FF + h];
  }
  wv[f] = s1;
  wv[FF + f] = s2;
}

__global__ __launch_bounds__(WAVES * 32) void gat_kernel(
    const float* __restrict__ item, const float* __restrict__ entity,
    const int* __restrict__ adj, const float* __restrict__ wv,
    float* __restrict__ out) {
  __shared__ float tile_s[WAVES][MM * STRIDE];
  __shared__ float ebuf_s[WAVES][MM];

  const int lane = threadIdx.x & 31;
  const int wid  = threadIdx.x >> 5;
  const int n    = blockIdx.x * WAVES + wid;
  if (n >= NTOT) return;                 // whole-wave uniform: EXEC stays full

  float* tile = tile_s[wid];
  float* ebuf = ebuf_s[wid];

  const int koff  = (lane < 16) ? 0 : 2; // K sub-offset for this lane half
  const int rhalf = lane & 15;

  // --- B fragments: wv2 chunk replicated across all 16 N columns ----------
  // 32-bit B 4x16 layout (mirrors documented A 16x4 / C-D pattern):
  // lanes 0-15: vgpr0=K0,vgpr1=K1 ; lanes 16-31: vgpr0=K2,vgpr1=K3.
  const float* wv2 = wv + FF;
  v2f bfrag[16];
#pragma unroll
  for (int kt = 0; kt < 16; ++kt) {
    bfrag[kt].x = wv2[kt * 4 + koff];
    bfrag[kt].y = wv2[kt * 4 + koff + 1];
  }

  // --- s1 = dot(item[n], wv1); keep item regs for the residual epilogue ---
  const float it0 = item[n * FF + lane];
  const float it1 = item[n * FF + lane + 32];
  float s1 = it0 * wv[lane] + it1 * wv[lane + 32];
#pragma unroll
  for (int off = 16; off >= 1; off >>= 1) s1 += __shfl_xor(s1, off, 32);

  // --- Phase 1: stream the 16 KB entity tile HBM -> LDS (padded rows) -----
  const float* gsrc = entity + (size_t)n * (MM * FF);
#pragma unroll
  for (int i = 0; i < 32; ++i) {
    const int idx = i * 32 + lane;       // float4 index within the tile
    const int row = idx >> 4;
    const int c4  = (idx & 15) * 4;
#if USE_ASYNC
    __builtin_amdgcn_global_load_async_to_lds_b128(
        (gv4i_p)(gsrc + idx * 4),
        (lv4i_p)(tile + row * STRIDE + c4),
        0, 0);
#else
    const v4f v = __builtin_nontemporal_load((const v4f*)(gsrc + idx * 4));
    *(v4f*)(tile + row * STRIDE + c4) = v;
#endif
  }
#if USE_ASYNC
  __builtin_amdgcn_s_wait_asynccnt(0);
  __asm__ volatile("" ::: "memory");
#endif

  // --- Phase 2: e = tile @ wv2 via V_WMMA_F32_16X16X4_F32 ------------------
  // A 16x4 f32 layout: lanes 0-15 hold (row, K=kt*4+{0,1}) in vgpr0/1,
  // lanes 16-31 hold (row, K=kt*4+{2,3}).
#pragma unroll
  for (int mt = 0; mt < 4; ++mt) {
    v8f c = {};
    const float* rp = tile + (mt * 16 + rhalf) * STRIDE + koff;
#pragma unroll
    for (int kt = 0; kt < 16; ++kt) {
      v2f a2;
      a2.x = rp[kt * 4];
      a2.y = rp[kt * 4 + 1];
      c = __builtin_amdgcn_wmma_f32_16x16x4_f32(
          false, a2, false, bfrag[kt], (short)0, c, false, false);
    }
    // C/D 16x16 f32: vgpr r -> row r (lanes 0-15) / row 8+r (lanes 16-31);
    // all N columns equal (B replicated), so lanes 0 and 16 publish rows.
    if (lane == 0) {
#pragma unroll
      for (int r = 0; r < 8; ++r) ebuf[mt * 16 + r] = c[r];
    } else if (lane == 16) {
#pragma unroll
      for (int r = 0; r < 8; ++r) ebuf[mt * 16 + 8 + r] = c[r];
    }
  }

  // --- Phase 3: leaky_relu + adjacency mask + softmax over m ---------------
  float e0 = ebuf[lane] + s1;
  float e1 = ebuf[lane + 32] + s1;
  e0 = (e0 > 0.f) ? e0 : LALPHA * e0;
  e1 = (e1 > 0.f) ? e1 : LALPHA * e1;
  const int ad0 = adj[n * MM + lane];
  const int ad1 = adj[n * MM + lane + 32];
  e0 = (ad0 > 0) ? e0 : NEG_BIG;
  e1 = (ad1 > 0) ? e1 : NEG_BIG;

  float mx = fmaxf(e0, e1);
#pragma unroll
  for (int off = 16; off >= 1; off >>= 1) mx = fmaxf(mx, __shfl_xor(mx, off, 32));
  const float x0 = __expf(e0 - mx);
  const float x1 = __expf(e1 - mx);
  float sm = x0 + x1;
#pragma unroll
  for (int off = 16; off >= 1; off >>= 1) sm += __shfl_xor(sm, off, 32);
  const float inv = 1.0f / sm;           // all-masked row -> uniform 1/64, matches ref
  ebuf[lane]      = x0 * inv;            // reuse ebuf as attention buffer
  ebuf[lane + 32] = x1 * inv;

  // --- Phase 4: out[n,f] = sum_m att[m]*tile[m,f] + item[n,f] --------------
  // Column reads: bank = (4m + f) % 64, conflict-free for f = lane / lane+32.
  float acc0 = 0.f, acc1 = 0.f;
#pragma unroll 8
  for (int m = 0; m < MM; ++m) {
    const float w = ebuf[m];
    acc0 += w * tile[m * STRIDE + lane];
    acc1 += w * tile[m * STRIDE + lane + 32];
  }
  __builtin_nontemporal_store(acc0 + it0, out + n * FF + lane);
  __builtin_nontemporal_store(acc1 + it1, out + n * FF + lane + 32);
}

extern "C" void kernel_launch(void* const* d_in, const int* in_sizes, int n_in,
                              void* d_out, int out_size, void* d_ws, size_t ws_size,
                              hipStream_t stream) {
  const float* item   = (const float*)d_in[0];   // (20000, 64)
  const float* entity = (const float*)d_in[1];   // (20000, 64, 64)
  const int*   adj    = (const int*)d_in[2];     // (20000, 64)
  const float* W      = (const float*)d_in[3];   // (64, 64)
  const float* a      = (const float*)d_in[4];   // (128, 1)
  float*       out    = (float*)d_out;           // (20000, 64)
  float*       wvbuf  = (float*)d_ws;            // 128 floats scratch

  wv_precompute<<<1, 64, 0, stream>>>(W, a, wvbuf);
  gat_kernel<<<(NTOT + WAVES - 1) / WAVES, WAVES * 32, 0, stream>>>(
      item, entity, adj, wvbuf, out);
}